// LDS_84104049590333
// MI455X (gfx1250) — compile-verified
//
#include <hip/hip_runtime.h>

// ---------------------------------------------------------------------------
// LDS (linear dynamical system) -> causal Toeplitz conv -> f32 WMMA GEMM
//
//   y[b,t] = sum_{d=0}^{t} kk[d]*x[b,t-d] + g[t]
//   kk[d]  = sum_s C[s]*Bp[s]*A[s]^d + (d<KX ? M[d] : 0)
//   g[t]   = sum_s C[s]*A[s]^(t+1)*h0[s]
//
// Y = X * U with U[tau,t] = kk[t-tau]: 896x1024x1024 f32 GEMM via
// V_WMMA_F32_16X16X4_F32, with B-fragments synthesized from kk, X tiles
// DMA'd into LDS by the Tensor Data Mover.
// ---------------------------------------------------------------------------

#define BSZ       896
#define SEQ_LEN   1024
#define STATE_DIM 256
#define KX        5

typedef __attribute__((ext_vector_type(2))) float        v2f;
typedef __attribute__((ext_vector_type(8))) float        v8f;
typedef __attribute__((ext_vector_type(4))) unsigned int u32x4;
typedef __attribute__((ext_vector_type(4))) int          i32x4;
typedef __attribute__((ext_vector_type(8))) int          i32x8;

// Workspace layout (floats):
//   ws[0 .. 2046]    : kkr (REVERSED kernel): kkr[1023-d] = kk[d]; kkr[i>1023] = 0
//   ws[2048 .. 3071] : g[t]
#define WS_KK_OFF 0
#define WS_G_OFF  2048

// ---------------------------------------------------------------------------
// Kernel 1: build kkr and g.  Grid: 16 blocks x 256 threads; block j owns
// d in [64j, 64j+64).  Thread s owns state s; A^(64j) by repeated squaring
// (f32 multiply chain == the reference's own error profile), then a 64-step
// recurrence fills an LDS tile which is reduced over states.
// ---------------------------------------------------------------------------
__global__ __launch_bounds__(256) void lds_build_kk(
    const float* __restrict__ A, const float* __restrict__ Bp,
    const float* __restrict__ C, const float* __restrict__ M,
    const float* __restrict__ h0, float* __restrict__ ws) {
  __shared__ float cont[64 * 257];   // [dd][s], stride 257 kills bank conflicts

  const int s  = threadIdx.x;        // state 0..255
  const int j  = blockIdx.x;         // d-chunk 0..15
  const int D0 = j * 64;

  const float a = A[s];
  float a64 = a;
#pragma unroll
  for (int i = 0; i < 6; ++i) a64 *= a64;      // a^64
  float aD0 = 1.0f;
  for (int q = 0; q < j; ++q) aD0 *= a64;      // a^(64j)

  // ---- pass 1: kk[d] = sum_s C*Bp*a^d, stored reversed ----
  float p = C[s] * Bp[s] * aD0;
  for (int dd = 0; dd < 64; ++dd) { cont[dd * 257 + s] = p; p *= a; }
  __syncthreads();
  if (s < 64) {
    float sum = 0.0f;
    for (int ss = 0; ss < STATE_DIM; ++ss) sum += cont[s * 257 + ss];
    const int d = D0 + s;
    ws[WS_KK_OFF + 1023 - d] = sum + (d < KX ? M[d] : 0.0f);
  }
  if (s >= 64 && s < 128) {                    // zero the acausal padding
    const int z = 1024 + D0 + (s - 64);
    if (z < 2047) ws[WS_KK_OFF + z] = 0.0f;
  }
  __syncthreads();

  // ---- pass 2: g[t] = sum_s C*h0*a^(t+1) ----
  float p2 = C[s] * h0[s] * a * aD0;
  for (int dd = 0; dd < 64; ++dd) { cont[dd * 257 + s] = p2; p2 *= a; }
  __syncthreads();
  if (s < 64) {
    float sum = 0.0f;
    for (int ss = 0; ss < STATE_DIM; ++ss) sum += cont[s * 257 + ss];
    ws[WS_G_OFF + D0 + s] = sum;
  }
}

// ---------------------------------------------------------------------------
// Kernel 2: Y = X * U, U[tau,t] = kk[t-tau] (upper-triangular Toeplitz).
// Block = 256 threads = 8 wave32; each wave owns a 16x16 output tile; block
// tile = 32 (batch) x 64 (time).  K-loop stops at tau < T0+64 (triangular
// skip).  X tiles arrive in LDS via TENSOR_LOAD_TO_LDS with hardware stride-36
// padding; B-fragments read ascending from the reversed kk in LDS.
// ---------------------------------------------------------------------------
#define KB 32

__global__ __launch_bounds__(256) void lds_toeplitz_wmma(
    const float* __restrict__ X, const float* __restrict__ ws,
    float* __restrict__ Y) {
  __shared__ float kkr[2048];        // reversed kk (2047 used)
  __shared__ float gs[64];           // g for this block's time columns
  __shared__ float xs[32 * 36];      // 32 rows x KB cols, padded stride 36

  const int tid  = threadIdx.x;
  const int lane = tid & 31;
  const int wv   = tid >> 5;         // wave 0..7
  const int wx   = wv & 3;           // time sub-tile 0..3
  const int wy   = wv >> 2;          // batch sub-tile 0..1
  const int T0   = blockIdx.x * 64;  // time-tile base
  const int B0   = blockIdx.y * 32;  // batch-tile base

  for (int i = tid; i < 2047; i += 256) kkr[i] = ws[WS_KK_OFF + i];
  if (tid < 64) gs[tid] = ws[WS_G_OFF + T0 + tid];

  // Fragment geometry (ISA 7.12.2, 32-bit operands, wave32):
  //   A 16x4: lane L -> M = L%16, holds K = 2*(L/16), 2*(L/16)+1
  //   B 4x16: lane L -> N = L%16, holds K = 2*(L/16), 2*(L/16)+1
  //   C/D 16x16: VGPR i -> M = i + 8*(L/16), N = L%16
  const int Mrow = lane & 15;
  const int K0   = (lane >> 4) << 1;
  const int tcol = T0 + wx * 16 + (lane & 15);

  v8f acc = {};
  const int kmax = T0 + 64;          // triangular bound

#if __has_builtin(__builtin_amdgcn_tensor_load_to_lds)
  // TDM descriptor (CDNA5 ISA ch.8): 2D f32 tile 32x32 from X -> xs.
  // LDS padding: pad_interval code 4 (= 32 dwords), pad_amount code 3
  // (= 4 dwords) reproduces the stride-36 layout in hardware.
  const unsigned lds_xs =
      (unsigned)(unsigned long long)(__attribute__((address_space(3))) float*)&xs[0];
  i32x8 g1;
  g1[0] = (2 << 16)                     // data_size = 4 bytes
        | (1 << 20)                     // pad_enable
        | (4 << 22)                     // pad_interval: 32 dwords
        | (3 << 25);                    // pad_amount:   4 dwords
  g1[1] = (SEQ_LEN & 0xFFFF) << 16;     // tensor_dim0 = 1024 (lo16)
  g1[2] = (BSZ & 0xFFFF) << 16;         // tensor_dim1 = 896  (lo16)
  g1[3] = (KB & 0xFFFF) << 16;          // tile_dim0 = 32
  g1[4] = KB;                           // tile_dim1 = 32, tile_dim2 = 0
  g1[5] = SEQ_LEN;                      // tensor_dim0_stride = 1024 (lo32)
  g1[6] = 0;
  g1[7] = 0;
  const i32x4 gz4 = {};
  const i32x8 gz8 = {};
  (void)gz8;
#endif

  const int ldr = tid >> 3;             // fallback-copy / prefetch row 0..31
  const int ldc = (tid & 7) * 4;        // fallback-copy / prefetch col

  for (int k0 = 0; k0 < kmax; k0 += KB) {
    __syncthreads();                    // previous tile fully consumed

#if __has_builtin(__builtin_amdgcn_tensor_load_to_lds)
    if (wv == 0) {                      // one wave drives the DMA
      const unsigned long long gaddr =
          (unsigned long long)(const void*)(X + (size_t)B0 * SEQ_LEN + k0);
      u32x4 g0;
      g0[0] = 1u;                                            // count=1 (user D#)
      g0[1] = lds_xs;                                        // LDS byte address
      g0[2] = (unsigned)(gaddr & 0xFFFFFFFFull);             // global addr lo
      g0[3] = (unsigned)((gaddr >> 32) & 0x01FFFFFFull)      // global addr hi
            | (2u << 30);                                    // type = 2 (image)
#if __clang_major__ >= 23
      __builtin_amdgcn_tensor_load_to_lds(g0, g1, gz4, gz4, gz8, 0);
#else
      __builtin_amdgcn_tensor_load_to_lds(g0, g1, gz4, gz4, 0);
#endif
    }
    __builtin_amdgcn_s_wait_tensorcnt(0);  // non-issuing waves: count==0, no-op
#else
    {
      const float4 v =
          *(const float4*)(X + (size_t)(B0 + ldr) * SEQ_LEN + k0 + ldc);
      *(float4*)&xs[ldr * 36 + ldc] = v;
    }
#endif
    if (k0 + KB < kmax) {               // WGP-scope prefetch of next X tile
      __builtin_prefetch(X + (size_t)(B0 + ldr) * SEQ_LEN + k0 + KB + ldc, 0, 3);
    }
    __syncthreads();                    // publish xs (and kkr/gs on iter 0)

#pragma unroll
    for (int kk0 = 0; kk0 < KB; kk0 += 4) {
      v2f af, bf;
      af.x = xs[(wy * 16 + Mrow) * 36 + kk0 + K0];
      af.y = xs[(wy * 16 + Mrow) * 36 + kk0 + K0 + 1];
      const int tau = k0 + kk0 + K0;
      // U[tau, t] = kkr[1023 - t + tau]; ascending pair -> clean 2addr load
      bf.x = kkr[1023 - tcol + tau];
      bf.y = kkr[1024 - tcol + tau];
      acc = __builtin_amdgcn_wmma_f32_16x16x4_f32(
          false, af, false, bf, (short)0, acc, false, false);
    }
  }

  // Epilogue: add g[t] and store.
  const float g   = gs[wx * 16 + (lane & 15)];
  const int rbase = B0 + wy * 16 + ((lane >> 4) << 3);
#pragma unroll
  for (int i = 0; i < 8; ++i) {
    Y[(size_t)(rbase + i) * SEQ_LEN + tcol] = acc[i] + g;
  }
}

// ---------------------------------------------------------------------------
extern "C" void kernel_launch(void* const* d_in, const int* in_sizes, int n_in,
                              void* d_out, int out_size, void* d_ws,
                              size_t ws_size, hipStream_t stream) {
  const float* x  = (const float*)d_in[0];  // (896,1024,1)
  const float* A  = (const float*)d_in[1];  // (256,)
  const float* Bp = (const float*)d_in[2];  // (1,256)
  const float* C  = (const float*)d_in[3];  // (256,1)
  const float* M  = (const float*)d_in[4];  // (1,1,5)
  const float* h0 = (const float*)d_in[5];  // (256,)
  float* out = (float*)d_out;
  float* ws  = (float*)d_ws;                // needs ~12.3 KB

  lds_build_kk<<<dim3(16), dim3(256), 0, stream>>>(A, Bp, C, M, h0, ws);

  dim3 grid(SEQ_LEN / 64, BSZ / 32);        // 16 x 28 blocks
  lds_toeplitz_wmma<<<grid, dim3(256), 0, stream>>>(x, ws, out);
}